// model1_75806172774608
// MI455X (gfx1250) — compile-verified
//
#include <hip/hip_runtime.h>
#include <math.h>

typedef float v2f __attribute__((ext_vector_type(2)));
typedef float v8f __attribute__((ext_vector_type(8)));

#define N_NODES 100000
#define N_EDGES 3200000

__device__ __forceinline__ float fast_sigmoid(float x) {
    // 1/(1+e^-x) with v_rcp_f32 (~1 ulp) instead of the IEEE div sequence
    return __builtin_amdgcn_rcpf(1.0f + __expf(-x));
}

// ---------------------------------------------------------------------------
// zero kernel (float4 granularity)
// ---------------------------------------------------------------------------
__global__ __launch_bounds__(256) void zero4_kernel(float4* __restrict__ p, int n4) {
    int i = blockIdx.x * blockDim.x + threadIdx.x;
    if (i < n4) p[i] = make_float4(0.f, 0.f, 0.f, 0.f);
}

// ---------------------------------------------------------------------------
// edge aggregation: agg[dst][f] += feat[src][f]   (agg stride fixed at 16)
// Vector loads of the source row, scalar f32 atomics into L2-resident agg.
// ---------------------------------------------------------------------------
template <int DIM, int STRIDE>
__global__ __launch_bounds__(256)
void agg_kernel(const int* __restrict__ src, const int* __restrict__ dst,
                const float* __restrict__ feat, float* __restrict__ agg,
                int n_edges) {
    int i = blockIdx.x * blockDim.x + threadIdx.x;
    if (i >= n_edges) return;
    int s = src[i];
    int d = dst[i];
    const float* fp = feat + (size_t)s * STRIDE;
    float* ap = agg + (size_t)d * 16;
    if (DIM == 4) {
        float4 a = *(const float4*)fp;
        atomicAdd(&ap[0], a.x);
        atomicAdd(&ap[1], a.y);
        atomicAdd(&ap[2], a.z);
        atomicAdd(&ap[3], a.w);
    } else {
        float4 a = *(const float4*)(fp + 0);
        float4 b = *(const float4*)(fp + 4);
        float2 c = *(const float2*)(fp + 8);
        atomicAdd(&ap[0], a.x);
        atomicAdd(&ap[1], a.y);
        atomicAdd(&ap[2], a.z);
        atomicAdd(&ap[3], a.w);
        atomicAdd(&ap[4], b.x);
        atomicAdd(&ap[5], b.y);
        atomicAdd(&ap[6], b.z);
        atomicAdd(&ap[7], b.w);
        atomicAdd(&ap[8], c.x);
        atomicAdd(&ap[9], c.y);
    }
}

// ---------------------------------------------------------------------------
// Per-16-node-tile MLP using V_WMMA_F32_16X16X4_F32 (one wave per block).
//   v      = agg + in_feat                       [16 x IN_DIM]
//   role   = sigmoid(v @ W^T + b)                [16 x 10] -> role_out (stride16)
//   out2   = softmax(role @ W2^T + b2, axis=-1)  [16 x 4]
//   f_partial[blk&63][0..3] += tile-sum of out2
//
// WMMA f32 16x16x4 VGPR layouts (ISA 7.12.2), lane L, m=n=L%16, h=L/16:
//   A (16x4): reg r = A[m][2h+r]   B (4x16): reg r = B[2h+r][n]
//   C/D:      reg j = D[j+8h][n]
// Weights are staged zero-padded in LDS so WMMA operand builds are plain
// unconditional ds loads (no exec manipulation near the WMMAs).
// ---------------------------------------------------------------------------
template <int IN_DIM, int IN_STRIDE>
__global__ __launch_bounds__(32)
void mlp_kernel(const float* __restrict__ in_feat,
                const float* __restrict__ agg,          // stride 16, pads zero
                const float* __restrict__ W,            // 10 x IN_DIM row-major
                const float* __restrict__ bvec,         // 10
                float* __restrict__ role_out,           // stride 16
                const float* __restrict__ W2,           // 4 x 10
                const float* __restrict__ b2,           // 4
                float* __restrict__ f_partial)          // 64 x 4
{
    __shared__ float lds_v[256];    // v tile,    16x16 padded
    __shared__ float lds_r[256];    // role tile, 16x16 padded
    __shared__ float lds_w[256];    // W  padded to 16x16 (row n, col k)
    __shared__ float lds_w2[256];   // W2 padded to 16x16
    __shared__ float lds_o[256];    // logits tile, 16x16 (cols 0..3 used)
    __shared__ float lds_b[16];
    __shared__ float lds_b2[16];

    const int lane  = threadIdx.x;  // 0..31
    const int m     = lane & 15;    // node row / output col
    const int h     = lane >> 4;    // 0 or 1
    const int node0 = blockIdx.x * 16;

    // ---- stage padded weights + biases into LDS (once per block) ----
    for (int i = lane; i < 256; i += 32) {
        int n_ = i >> 4, k_ = i & 15;
        lds_w[i]  = (n_ < 10 && k_ < IN_DIM) ? W[n_ * IN_DIM + k_] : 0.0f;
        lds_w2[i] = (n_ < 4 && k_ < 10)      ? W2[n_ * 10 + k_]    : 0.0f;
    }
    if (lane < 16) {
        lds_b[lane]  = (lane < 10) ? bvec[lane] : 0.0f;
        lds_b2[lane] = (lane < 4)  ? b2[lane]   : 0.0f;
    }

    // ---- build v = agg + in_feat tile in LDS (float4 granularity) ----
    for (int i = lane; i < 64; i += 32) {
        int nd = i >> 2, fq = i & 3;
        int g = node0 + nd;
        float4 av = *(const float4*)(agg + (size_t)g * 16 + fq * 4);
        float4 vv;
        if (IN_STRIDE == 16) {
            // role buffer: all 16 cols valid (pads stored as zero)
            float4 fv = *(const float4*)(in_feat + (size_t)g * 16 + fq * 4);
            vv = make_float4(av.x + fv.x, av.y + fv.y, av.z + fv.z, av.w + fv.w);
        } else {
            if (fq == 0) {
                float4 fv = *(const float4*)(in_feat + (size_t)g * 4);
                vv = make_float4(av.x + fv.x, av.y + fv.y, av.z + fv.z, av.w + fv.w);
            } else {
                vv = make_float4(0.f, 0.f, 0.f, 0.f);  // agg pad cols are zero
            }
        }
        *(float4*)(lds_v + i * 4) = vv;
    }
    __syncthreads();

    // ---- matmul 1: role_pre = v @ W^T (K padded to 4*NCHUNK) ----
    constexpr int NCHUNK = (IN_DIM + 3) / 4;  // 1 or 3
    const int n = m;
    v8f acc = {0.f, 0.f, 0.f, 0.f, 0.f, 0.f, 0.f, 0.f};
#pragma unroll
    for (int c = 0; c < NCHUNK; ++c) {
        int k0 = 4 * c + 2 * h;
        v2f a, b;
        a.x = lds_v[m * 16 + k0];
        a.y = lds_v[m * 16 + k0 + 1];
        b.x = lds_w[n * 16 + k0];
        b.y = lds_w[n * 16 + k0 + 1];
        acc = __builtin_amdgcn_wmma_f32_16x16x4_f32(false, a, false, b,
                                                    (short)0, acc, false, false);
    }

    // ---- bias + sigmoid -> lds_r (zero-padded cols 10..15) ----
    // Compute unconditionally, then select: v_cndmask, no exec branches.
    float bn   = lds_b[n];
    float keep = (n < 10) ? 1.0f : 0.0f;
#pragma unroll
    for (int j = 0; j < 8; ++j) {
        int nd = j + 8 * h;
        float sg = fast_sigmoid(acc[j] + bn);
        lds_r[nd * 16 + n] = sg * keep;
    }
    __syncthreads();

    // ---- coalesced role tile writeback (b128 stores) ----
    for (int i = lane; i < 64; i += 32) {
        *(float4*)(role_out + (size_t)node0 * 16 + i * 4) =
            *(const float4*)(lds_r + i * 4);
    }

    // ---- matmul 2: out2_pre = role @ W2^T (K = 10 padded to 12) ----
    v8f acc2 = {0.f, 0.f, 0.f, 0.f, 0.f, 0.f, 0.f, 0.f};
#pragma unroll
    for (int c = 0; c < 3; ++c) {
        int k0 = 4 * c + 2 * h;
        v2f a, b;
        a.x = lds_r[m * 16 + k0];
        a.y = lds_r[m * 16 + k0 + 1];
        b.x = lds_w2[n * 16 + k0];
        b.y = lds_w2[n * 16 + k0 + 1];
        acc2 = __builtin_amdgcn_wmma_f32_16x16x4_f32(false, a, false, b,
                                                     (short)0, acc2, false, false);
    }

    // ---- pre-softmax logits to LDS (unconditional, 16-col stride) ----
    float b2n = lds_b2[n];
#pragma unroll
    for (int j = 0; j < 8; ++j) {
        int nd = j + 8 * h;
        lds_o[nd * 16 + n] = acc2[j] + b2n;
    }
    __syncthreads();

    // ---- per-node softmax (one lane per node, cols 0..3) ----
    if (lane < 16) {
        float x0 = lds_o[lane * 16 + 0];
        float x1 = lds_o[lane * 16 + 1];
        float x2 = lds_o[lane * 16 + 2];
        float x3 = lds_o[lane * 16 + 3];
        float mx = fmaxf(fmaxf(x0, x1), fmaxf(x2, x3));
        float e0 = __expf(x0 - mx), e1 = __expf(x1 - mx);
        float e2 = __expf(x2 - mx), e3 = __expf(x3 - mx);
        float inv = __builtin_amdgcn_rcpf(e0 + e1 + e2 + e3);
        lds_o[lane * 16 + 0] = e0 * inv;
        lds_o[lane * 16 + 1] = e1 * inv;
        lds_o[lane * 16 + 2] = e2 * inv;
        lds_o[lane * 16 + 3] = e3 * inv;
    }
    __syncthreads();

    // ---- tile reduction of out2 into spread f_final accumulator ----
    if (lane < 4) {
        float s = 0.0f;
#pragma unroll
        for (int nd = 0; nd < 16; ++nd) s += lds_o[nd * 16 + lane];
        atomicAdd(&f_partial[(blockIdx.x & 63) * 4 + lane], s);
    }
}

// ---------------------------------------------------------------------------
// Head: finish f_final reduction + tiny MLP head + loss (single thread).
// ---------------------------------------------------------------------------
__global__ __launch_bounds__(32)
void head_kernel(const float* __restrict__ f_partial,   // 64 x 4
                 const float* __restrict__ gf,          // 5
                 const int* __restrict__ label,
                 const float* __restrict__ W3,  const float* __restrict__ b3,
                 const float* __restrict__ W4a, const float* __restrict__ b4a,
                 const float* __restrict__ W4b, const float* __restrict__ b4b,
                 float* __restrict__ out)
{
    if (threadIdx.x != 0 || blockIdx.x != 0) return;

    float ff[4] = {0.f, 0.f, 0.f, 0.f};
    for (int s = 0; s < 64; ++s)
        for (int f = 0; f < 4; ++f)
            ff[f] += f_partial[s * 4 + f];

    float o3[3];
    for (int j = 0; j < 3; ++j) {
        float a = b3[j];
        for (int k = 0; k < 5; ++k) a += gf[k] * W3[j * 5 + k];
        o3[j] = fast_sigmoid(a);
    }

    float o4[7] = {ff[0], ff[1], ff[2], ff[3], o3[0], o3[1], o3[2]};

    float hv[4];
    for (int j = 0; j < 4; ++j) {
        float a = b4a[j];
        for (int k = 0; k < 7; ++k) a += o4[k] * W4a[j * 7 + k];
        hv[j] = fast_sigmoid(a);
    }

    float lg[3];
    for (int j = 0; j < 3; ++j) {
        float a = b4b[j];
        for (int k = 0; k < 4; ++k) a += hv[k] * W4b[j * 4 + k];
        lg[j] = a;
    }

    // out5 = softmax(lg)
    float mx = fmaxf(lg[0], fmaxf(lg[1], lg[2]));
    float e0 = __expf(lg[0] - mx), e1 = __expf(lg[1] - mx), e2 = __expf(lg[2] - mx);
    float inv = __builtin_amdgcn_rcpf(e0 + e1 + e2);
    float o5[3] = {e0 * inv, e1 * inv, e2 * inv};

    // loss = -(log_softmax(out5))[label]
    float mx2 = fmaxf(o5[0], fmaxf(o5[1], o5[2]));
    float s2 = __expf(o5[0] - mx2) + __expf(o5[1] - mx2) + __expf(o5[2] - mx2);
    float lse = mx2 + __logf(s2);
    int lb = *label;
    if (lb < 0) lb = 0;
    if (lb > 2) lb = 2;
    out[0] = lse - o5[lb];
}

// ---------------------------------------------------------------------------
// Launch
// ---------------------------------------------------------------------------
extern "C" void kernel_launch(void* const* d_in, const int* in_sizes, int n_in,
                              void* d_out, int out_size, void* d_ws, size_t ws_size,
                              hipStream_t stream) {
    const float* x     = (const float*)d_in[0];
    const int*   esrc  = (const int*)d_in[1];
    const int*   edst  = (const int*)d_in[2];
    const float* gf    = (const float*)d_in[3];
    const int*   label = (const int*)d_in[4];
    const float* W1  = (const float*)d_in[5];
    const float* b1  = (const float*)d_in[6];
    const float* W11 = (const float*)d_in[7];
    const float* b11 = (const float*)d_in[8];
    const float* W2  = (const float*)d_in[9];
    const float* b2  = (const float*)d_in[10];
    const float* W3  = (const float*)d_in[11];
    const float* b3  = (const float*)d_in[12];
    const float* W4a = (const float*)d_in[13];
    const float* b4a = (const float*)d_in[14];
    const float* W4b = (const float*)d_in[15];
    const float* b4b = (const float*)d_in[16];

    float* ws    = (float*)d_ws;
    float* agg   = ws;                              // N_NODES * 16
    float* role  = ws + (size_t)N_NODES * 16;       // N_NODES * 16
    float* fpart = ws + (size_t)2 * N_NODES * 16;   // 64 * 4

    const int AGG_Q = N_NODES * 16 / 4;             // float4 count
    const int zgrid = (AGG_Q + 255) / 256;
    const int egrid = (N_EDGES + 255) / 256;
    const int mgrid = N_NODES / 16;                 // 6250 exactly

    zero4_kernel<<<1, 256, 0, stream>>>((float4*)fpart, 64);

    // Layer 0: aggregate raw x (dim 4), 4->10 MLP
    zero4_kernel<<<zgrid, 256, 0, stream>>>((float4*)agg, AGG_Q);
    agg_kernel<4, 4><<<egrid, 256, 0, stream>>>(esrc, edst, x, agg, N_EDGES);
    mlp_kernel<4, 4><<<mgrid, 32, 0, stream>>>(x, agg, W1, b1, role, W2, b2, fpart);

    // Layers 1..3: aggregate role (dim 10), 10->10 MLP (role updated in place)
    for (int l = 1; l < 4; ++l) {
        zero4_kernel<<<zgrid, 256, 0, stream>>>((float4*)agg, AGG_Q);
        agg_kernel<10, 16><<<egrid, 256, 0, stream>>>(esrc, edst, role, agg, N_EDGES);
        mlp_kernel<10, 16><<<mgrid, 32, 0, stream>>>(role, agg, W11, b11, role, W2,
                                                     b2, fpart);
    }

    head_kernel<<<1, 32, 0, stream>>>(fpart, gf, label, W3, b3, W4a, b4a, W4b, b4b,
                                      (float*)d_out);
}